// LSTMModel_61804579389720
// MI455X (gfx1250) — compile-verified
//
#include <hip/hip_runtime.h>
#include <math.h>

typedef float v2f __attribute__((ext_vector_type(2)));
typedef float v8f __attribute__((ext_vector_type(8)));

// Problem constants
#define BB 16      // batch
#define TT 2048    // time steps
#define FF 512     // features (= K of both GEMMs)
#define GG 2048    // 4*FF gate width (= N)
#define NWG 16     // persistent workgroups in scan kernel

// ---------------------------------------------------------------------------
// Phase 1: xg = X @ Wi + b      X:[32768,512]  Wi:[512,2048]  xg:[32768,2048]
// Block: 256 threads (8 waves). Each block: one 16-row M tile x 128 columns.
// Each wave computes one 16x16 C tile with K=512 via 128 V_WMMA_F32_16X16X4_F32.
// A tile is staged in LDS in WMMA A-fragment layout and shared by all 8 waves.
// ---------------------------------------------------------------------------
__global__ void __launch_bounds__(256) xg_gemm(const float* __restrict__ X,
                                               const float* __restrict__ Wi,
                                               const float* __restrict__ bias,
                                               float* __restrict__ XG)
{
    __shared__ float aLds[128 * 32 * 2];   // 32 KB: [frag f][lane][2]

    const int wave = threadIdx.x >> 5;
    const int lane = threadIdx.x & 31;
    const int row0 = blockIdx.y * 16;                 // M tile base (0..32767)
    const int n0   = blockIdx.x * 128 + wave * 16;    // N tile base (0..2047)

    // Stage A tile (16 rows x 512 K) into fragment-swizzled LDS.
    // Fragment f covers K = 4f..4f+3. Lane l<16 holds (M=l, K=4f,4f+1),
    // lane l>=16 holds (M=l-16, K=4f+2,4f+3). Two floats contiguous per lane.
    for (int idx = threadIdx.x; idx < 128 * 32; idx += 256) {
        const int f = idx >> 5;
        const int l = idx & 31;
        const int r = l & 15;
        const int kk = 4 * f + ((l >> 4) ? 2 : 0);
        const float* p = X + (size_t)(row0 + r) * FF + kk;
        aLds[idx * 2 + 0] = p[0];
        aLds[idx * 2 + 1] = p[1];
    }
    __syncthreads();

    const int bl   = lane & 15;            // N column within tile
    const int koff = (lane >> 4) ? 2 : 0;  // K sub-offset for this half-wave

    v8f acc = {};
#pragma unroll 4
    for (int f = 0; f < 128; ++f) {
        const int k = 4 * f;
        v2f a = *(const v2f*)&aLds[(f * 32 + lane) * 2];
        v2f b;
        const float* wp = Wi + (size_t)(k + koff) * GG + n0 + bl;
        b.x = wp[0];
        b.y = wp[GG];
        acc = __builtin_amdgcn_wmma_f32_16x16x4_f32(
            false, a, false, b, (short)0, acc, false, false);
    }

    const float bb = bias[n0 + bl];
    const int rbase = (lane >> 4) ? 8 : 0;
#pragma unroll
    for (int r = 0; r < 8; ++r) {
        XG[(size_t)(row0 + rbase + r) * GG + n0 + bl] = acc[r] + bb;
    }
}

// ---------------------------------------------------------------------------
// Phase 2: persistent recurrent scan. 16 workgroups x 256 threads.
// WG j owns h columns [j*32, j*32+32) and the matching 4x32 gate columns.
// Wave w (gate = w>>1, half = w&1) owns z columns gate*512 + j*32 + half*16.
// Wh slice (512 x 128 fp32 = 256 KB) lives in LDS, fragment-swizzled, for the
// entire 2048-step scan. c state lives in LDS. h is exchanged between WGs via
// a double-buffered global buffer + atomic grid barrier each step.
// ---------------------------------------------------------------------------
__global__ void __launch_bounds__(256) lstm_scan(
    const float* __restrict__ XG,     // [32768, 2048], row = b*T + t
    const float* __restrict__ Wh,     // [512, 2048]
    float* __restrict__ Y,            // [B, T, F] output sequence of this layer
    float* __restrict__ hbuf,         // [2][16][512] double-buffered h
    float* __restrict__ cOut,         // [16*512] final c for this layer
    float* __restrict__ hOut,         // [16*512] final h for this layer
    unsigned* __restrict__ gbar)      // grid barrier counter (zeroed by host)
{
    extern __shared__ float smem[];
    float* whLds = smem;                    // 8*128*32*2 = 65536 f (256 KB)
    float* hLds  = whLds + 65536;           // 128*32*2   =  8192 f ( 32 KB)
    float* zLds  = hLds + 8192;             // 16*128     =  2048 f (  8 KB)
    float* cLds  = zLds + 2048;             // 16*32      =   512 f (  2 KB)

    const int wave = threadIdx.x >> 5;
    const int lane = threadIdx.x & 31;
    const int wg   = blockIdx.x;            // 0..15
    const int gate = wave >> 1;             // 0..3 (i,f,g,o)
    const int half = wave & 1;
    const int n0   = gate * FF + wg * 32 + half * 16; // global z column base
    const int bl   = lane & 15;
    const int kq   = (lane >> 4) ? 2 : 0;

    // Preload this wave's Wh slice into fragment-swizzled LDS (reused 2048x).
    for (int f = 0; f < 128; ++f) {
        const float* wp = Wh + (size_t)(4 * f + kq) * GG + n0 + bl;
        float* dst = &whLds[((wave * 128 + f) * 32 + lane) * 2];
        dst[0] = wp[0];
        dst[1] = wp[GG];
    }
    for (int i = threadIdx.x; i < 512; i += 256) cLds[i] = 0.0f;
    __syncthreads();

    for (int t = 0; t < TT; ++t) {
        // Stage h(t) from global into A-fragment-swizzled LDS.
        const float* hg = hbuf + (size_t)(t & 1) * (BB * FF);
        for (int idx = threadIdx.x; idx < 128 * 32; idx += 256) {
            const int f = idx >> 5;
            const int l = idx & 31;
            const int r = l & 15;
            const int kk = 4 * f + ((l >> 4) ? 2 : 0);
            const float* p = hg + r * FF + kk;
            hLds[idx * 2 + 0] = p[0];
            hLds[idx * 2 + 1] = p[1];
        }
        __syncthreads();

        // z_tile = h @ Wh_slice  (16x16, K=512)
        v8f acc = {};
#pragma unroll 4
        for (int f = 0; f < 128; ++f) {
            v2f a = *(const v2f*)&hLds[(f * 32 + lane) * 2];
            v2f b = *(const v2f*)&whLds[((wave * 128 + f) * 32 + lane) * 2];
            acc = __builtin_amdgcn_wmma_f32_16x16x4_f32(
                false, a, false, b, (short)0, acc, false, false);
        }

        // Scatter C tile into zLds[16][128] (cols grouped by gate).
        {
            const int rb = (lane >> 4) ? 8 : 0;
            const int zc = gate * 32 + half * 16 + bl;
#pragma unroll
            for (int r = 0; r < 8; ++r) zLds[(rb + r) * 128 + zc] = acc[r];
        }
        __syncthreads();

        // Elementwise gate math for our 16 x 32 slice of h/c.
        float* hw = hbuf + (size_t)((t + 1) & 1) * (BB * FF);
        for (int idx = threadIdx.x; idx < 512; idx += 256) {
            const int m = idx >> 5;            // batch row
            const int n = idx & 31;            // local h column
            const int colh = wg * 32 + n;      // global h column
            const size_t xrow = ((size_t)m * TT + t) * GG;
            const float zi = zLds[m * 128 + 0 * 32 + n] + XG[xrow + 0 * FF + colh];
            const float zf = zLds[m * 128 + 1 * 32 + n] + XG[xrow + 1 * FF + colh];
            const float zg = zLds[m * 128 + 2 * 32 + n] + XG[xrow + 2 * FF + colh];
            const float zo = zLds[m * 128 + 3 * 32 + n] + XG[xrow + 3 * FF + colh];
            const float si = 1.0f / (1.0f + __expf(-zi));
            const float sf = 1.0f / (1.0f + __expf(-zf));
            const float so = 1.0f / (1.0f + __expf(-zo));
            const float tg = tanhf(zg);
            const float c  = sf * cLds[idx] + si * tg;
            cLds[idx] = c;
            const float h = so * tanhf(c);
            hw[m * FF + colh] = h;
            Y[((size_t)m * TT + t) * FF + colh] = h;
            if (t == TT - 1) {
                cOut[m * FF + colh] = c;
                hOut[m * FF + colh] = h;
            }
        }
        __syncthreads();

        // Grid barrier across the 16 persistent workgroups (monotonic counter).
        __threadfence();
        if (threadIdx.x == 0) {
            __hip_atomic_fetch_add(gbar, 1u, __ATOMIC_RELEASE,
                                   __HIP_MEMORY_SCOPE_AGENT);
            const unsigned target = (unsigned)NWG * (unsigned)(t + 1);
            while (__hip_atomic_load(gbar, __ATOMIC_ACQUIRE,
                                     __HIP_MEMORY_SCOPE_AGENT) < target) {
                __builtin_amdgcn_s_sleep(1);
            }
        }
        __syncthreads();
    }
}

// ---------------------------------------------------------------------------
// Host-side orchestration (graph-capture safe: only kernel launches + async
// memset on `stream`).
// ---------------------------------------------------------------------------
extern "C" void kernel_launch(void* const* d_in, const int* in_sizes, int n_in,
                              void* d_out, int out_size, void* d_ws, size_t ws_size,
                              hipStream_t stream) {
    const float* x    = (const float*)d_in[0];   // [16,2048,512]
    const float* Wi   = (const float*)d_in[1];   // [2,512,2048]
    const float* Wh   = (const float*)d_in[2];   // [2,512,2048]
    const float* bias = (const float*)d_in[3];   // [2,2048]

    float* out   = (float*)d_out;
    float* yTop  = out;                                   // [16,2048,512]
    float* csOut = out + (size_t)BB * TT * FF;            // [2,16,512]
    float* hsOut = csOut + (size_t)2 * BB * FF;           // [2,16,512]

    float* ws  = (float*)d_ws;
    float* XG  = ws;                                      // 32768*2048 f (256 MB)
    float* Y0  = XG + (size_t)BB * TT * GG;               // 16*2048*512 f (64 MB)
    float* hb  = Y0 + (size_t)BB * TT * FF;               // 2*16*512 f
    unsigned* cnt = (unsigned*)(hb + 2 * BB * FF);

    const size_t smemBytes = (size_t)(65536 + 8192 + 2048 + 512) * sizeof(float);

    for (int l = 0; l < 2; ++l) {
        // zero h double-buffer + barrier counter (contiguous)
        hipMemsetAsync(hb, 0, (size_t)2 * BB * FF * sizeof(float) + 64, stream);

        const float* Xin = (l == 0) ? x : Y0;
        float*       Yo  = (l == 0) ? Y0 : yTop;

        xg_gemm<<<dim3(GG / 128, (BB * TT) / 16), 256, 0, stream>>>(
            Xin, Wi + (size_t)l * FF * GG, bias + (size_t)l * GG, XG);

        lstm_scan<<<dim3(NWG), 256, smemBytes, stream>>>(
            XG, Wh + (size_t)l * FF * GG, Yo, hb,
            csOut + (size_t)l * BB * FF, hsOut + (size_t)l * BB * FF, cnt);
    }
}